// SparseMicroRefine_67190468379263
// MI455X (gfx1250) — compile-verified
//
#include <hip/hip_runtime.h>
#include <hip/hip_bf16.h>
#include <stdint.h>
#include <math.h>

typedef float v4f __attribute__((ext_vector_type(4)));

#define NBUF    4
#define MAX_TPR 1024   // max threads per row = D/4 (D up to 4096)

__device__ __forceinline__ float refine_elem(float v, float W1, float B1, float W2, float B2) {
    // silu(silu(v*w1+b1)*w2+b2), full-accuracy exp (kernel is bandwidth-bound)
    float h = __builtin_fmaf(v, W1, B1);
    h = h / (1.0f + expf(-h));
    float g = __builtin_fmaf(h, W2, B2);
    return g / (1.0f + expf(-g));
}

// ---------------------------------------------------------------------------
// Kernel 1: exact top-k mask with jax.lax.top_k tie-breaking (lower index wins)
// ---------------------------------------------------------------------------
__global__ void topk_mask_kernel(const float* __restrict__ imp,
                                 unsigned char* __restrict__ mask,
                                 int D, int keep) {
    int c = blockIdx.x * blockDim.x + threadIdx.x;
    if (c >= D) return;
    float v = imp[c];
    int rank = 0;
    for (int j = 0; j < D; ++j) {
        float u = imp[j];                 // uniform address per iter -> cached/broadcast
        rank += ((u > v) || (u == v && j < c)) ? 1 : 0;
    }
    mask[c] = (rank < keep) ? 1u : 0u;
}

// ---------------------------------------------------------------------------
// Kernel 2: streaming refine with CDNA5 async global->LDS 4-deep pipeline.
// blockDim.x == D/4; thread t owns channels [4t, 4t+3] of every row it touches.
// Each thread consumes exactly the LDS bytes it async-loaded, so per-wave
// ASYNCcnt waits are sufficient (no workgroup barriers). Both the read and
// write streams are single-use (537MB vs 192MB L2) -> non-temporal both ways.
// ---------------------------------------------------------------------------
__global__ void refine_pipe_kernel(const float* __restrict__ x,
                                   float* __restrict__ out,
                                   const unsigned char* __restrict__ mask,
                                   const float* __restrict__ w1p,
                                   const float* __restrict__ b1p,
                                   const float* __restrict__ w2p,
                                   const float* __restrict__ b2p,
                                   int rows, int D) {
    __shared__ v4f sbuf[NBUF * MAX_TPR];          // 64 KB ring (of 320 KB WGP LDS)

    const int tid = threadIdx.x;
    const int tpr = blockDim.x;                   // = D/4
    const float W1 = w1p[0], B1 = b1p[0], W2 = w2p[0], B2 = b2p[0];

    // per-thread channel mask, loaded once (channels fixed per thread)
    const uint32_t m4 = *(const uint32_t*)(mask + 4 * tid);
    const bool m0 = (m4 & 0x000000ffu) != 0;
    const bool m1 = (m4 & 0x0000ff00u) != 0;
    const bool m2 = (m4 & 0x00ff0000u) != 0;
    const bool m3 = (m4 & 0xff000000u) != 0;

    // absolute LDS byte address of the ring buffer
    typedef __attribute__((address_space(3))) unsigned char lds_u8;
    const uint32_t lds_base = (uint32_t)(uint64_t)(lds_u8*)(&sbuf[0]);

    const int rstep   = gridDim.x;
    const int rowB    = D * 4;                    // row pitch in bytes (16 KB)
    const int laneOff = tid * 16;

    // ---- prologue: fill the NBUF-deep pipeline (clamp past-end rows to row 0
    //      so the outstanding ASYNCcnt is always exactly NBUF) ----
    int pr = blockIdx.x;
#pragma unroll
    for (int k = 0; k < NBUF; ++k) {
        const int rr = (pr < rows) ? pr : 0;
        const uint32_t la = lds_base + (uint32_t)(k * tpr + tid) * 16u;
        const int go = rr * rowB + laneOff;
        asm volatile("global_load_async_to_lds_b128 %0, %1, %2 th:TH_LOAD_NT"
                     :: "v"(la), "v"(go), "s"(x)
                     : "memory");
        pr += rstep;
    }

    // ---- steady state ----
    int buf = 0;
    for (int r = blockIdx.x; r < rows; r += rstep) {
        // oldest of the NBUF outstanding async loads has landed in LDS
        // (async "done" returns in order -> waiting <=3 retires the oldest)
        asm volatile("s_wait_asynccnt 3" ::: "memory");

        v4f v = sbuf[buf * tpr + tid];            // ds_load_b128

        v4f o;
        o.x = m0 ? refine_elem(v.x, W1, B1, W2, B2) : v.x;
        o.y = m1 ? refine_elem(v.y, W1, B1, W2, B2) : v.y;
        o.z = m2 ? refine_elem(v.z, W1, B1, W2, B2) : v.z;
        o.w = m3 ? refine_elem(v.w, W1, B1, W2, B2) : v.w;

        // streaming store: don't let the 268MB write stream thrash L2
        __builtin_nontemporal_store(o, (v4f*)(out + (size_t)r * (size_t)D) + tid);

        // WAR: make sure our ds_load of this ring slot retired before refilling it
        asm volatile("s_wait_dscnt 0" ::: "memory");

        const int nr = r + NBUF * rstep;
        const int rr = (nr < rows) ? nr : 0;      // clamped dummy keeps count == NBUF
        const uint32_t la = lds_base + (uint32_t)(buf * tpr + tid) * 16u;
        const int go = rr * rowB + laneOff;
        asm volatile("global_load_async_to_lds_b128 %0, %1, %2 th:TH_LOAD_NT"
                     :: "v"(la), "v"(go), "s"(x)
                     : "memory");

        buf = (buf + 1) & (NBUF - 1);
    }
    // leftover dummy asyncs are drained by S_ENDPGM's implicit wait-idle
}

// ---------------------------------------------------------------------------
// Generic fallback (any D): plain grid-stride elementwise
// ---------------------------------------------------------------------------
__global__ void refine_simple_kernel(const float* __restrict__ x,
                                     float* __restrict__ out,
                                     const unsigned char* __restrict__ mask,
                                     const float* __restrict__ w1p,
                                     const float* __restrict__ b1p,
                                     const float* __restrict__ w2p,
                                     const float* __restrict__ b2p,
                                     long long N, int D) {
    const float W1 = w1p[0], B1 = b1p[0], W2 = w2p[0], B2 = b2p[0];
    long long i = (long long)blockIdx.x * blockDim.x + threadIdx.x;
    const long long stride = (long long)gridDim.x * blockDim.x;
    for (; i < N; i += stride) {
        float v = x[i];
        int c = (int)(i % (long long)D);
        out[i] = mask[c] ? refine_elem(v, W1, B1, W2, B2) : v;
    }
}

extern "C" void kernel_launch(void* const* d_in, const int* in_sizes, int n_in,
                              void* d_out, int out_size, void* d_ws, size_t ws_size,
                              hipStream_t stream) {
    const float* x   = (const float*)d_in[0];
    const float* imp = (const float*)d_in[1];
    const float* w1  = (const float*)d_in[2];
    const float* b1  = (const float*)d_in[3];
    const float* w2  = (const float*)d_in[4];
    const float* b2  = (const float*)d_in[5];
    float* out = (float*)d_out;

    const int D = in_sizes[1];
    const long long N = (long long)in_sizes[0];
    int keep = (int)((double)D * 0.25);
    if (keep < 1) keep = 1;

    unsigned char* mask = (unsigned char*)d_ws;

    {
        const int t = 256;
        const int g = (D + t - 1) / t;
        topk_mask_kernel<<<g, t, 0, stream>>>(imp, mask, D, keep);
    }

    if ((D % 4) == 0 && (D / 4) <= MAX_TPR && (N % (long long)D) == 0) {
        const int rows = (int)(N / (long long)D);
        const int tpr  = D / 4;                    // 1024 for D=4096
        int grid = 2048;                           // ~8 rows/block -> pipeline pays off
        if (grid > rows) grid = rows;
        if (grid < 1) grid = 1;
        refine_pipe_kernel<<<grid, tpr, 0, stream>>>(x, out, mask, w1, b1, w2, b2, rows, D);
    } else {
        const int t = 256;
        long long blocks = (N + t - 1) / t;
        int g = (blocks > 32768) ? 32768 : (int)blocks;
        if (g < 1) g = 1;
        refine_simple_kernel<<<g, t, 0, stream>>>(x, out, mask, w1, b1, w2, b2, N, D);
    }
}